// Vtrace_24249385353346
// MI455X (gfx1250) — compile-verified
//
#include <hip/hip_runtime.h>
#include <hip/hip_bf16.h>

#define T_LEN   524288
#define HIDDEN  256
#define GAMMA   0.98f

#define SCAN_BLOCKS  256
#define SCAN_THREADS 256
#define SEQ          8        // 256*256*8 == T_LEN

typedef __attribute__((ext_vector_type(2))) float v2f;
typedef __attribute__((ext_vector_type(8))) float v8f;

// ---------------------------------------------------------------------------
// Kernel 1: fused backbone. Each wave owns 16 timesteps (M=16). For each of
// 16 N-tiles of the 256 hidden units it runs V_WMMA_F32_16X16X4_F32 for s and
// s_prime (A = 16x4 state tile, B = 4x16 slice of W1^T, C = bias broadcast),
// applies relu in-register, and immediately reduces against the head weights
// (Wv, Wpi rows) so h is never written to memory. Epilogue computes softmax,
// rho, delta per timestep and stores the scan coefficients a=gamma*c, d=delta.
// ---------------------------------------------------------------------------
__global__ __launch_bounds__(256) void vtrace_backbone_wmma(
    const float* __restrict__ s, const int* __restrict__ act,
    const float* __restrict__ rew, const float* __restrict__ s_prime,
    const float* __restrict__ done, const float* __restrict__ mu,
    const float* __restrict__ W1, const float* __restrict__ b1,
    const float* __restrict__ Wpi, const float* __restrict__ bpi,
    const float* __restrict__ Wv, const float* __restrict__ bv,
    float* __restrict__ out, float* __restrict__ a_arr,
    float* __restrict__ d_arr, float* __restrict__ v_arr,
    float* __restrict__ vp_arr)
{
  const int wave = threadIdx.x >> 5;
  const int lane = threadIdx.x & 31;
  const int half = lane >> 4;     // 0: K=0..1 / rows 0..7 ; 1: K=2..3 / rows 8..15
  const int mrow = lane & 15;
  const int tile = (blockIdx.x * 8 + wave) * 16;

  // A-matrix (16x4 f32): lane holds s[tile+mrow][2*half + {0,1}]
  v2f aS, aP;
  {
    const float* p = s       + (size_t)(tile + mrow) * 4 + half * 2;
    const float* q = s_prime + (size_t)(tile + mrow) * 4 + half * 2;
    aS.x = p[0]; aS.y = p[1];
    aP.x = q[0]; aP.y = q[1];
  }

  float accv[8], accp0[8], accp1[8], accvp[8];
#pragma unroll
  for (int r = 0; r < 8; ++r) { accv[r] = 0.f; accp0[r] = 0.f; accp1[r] = 0.f; accvp[r] = 0.f; }

#pragma unroll
  for (int nt = 0; nt < 16; ++nt) {
    const int col = nt * 16 + mrow;             // hidden unit handled by this lane
    // B-matrix (4x16 f32) = W1^T slice: reg r, half h -> K = 2h+r -> W1[col][2h+r]
    v2f bmat;
    bmat.x = W1[col * 4 + half * 2 + 0];
    bmat.y = W1[col * 4 + half * 2 + 1];
    const float bias = b1[col];
    v8f c;
#pragma unroll
    for (int r = 0; r < 8; ++r) c[r] = bias;

    v8f h  = __builtin_amdgcn_wmma_f32_16x16x4_f32(false, aS, false, bmat, (short)0, c, false, false);
    v8f hp = __builtin_amdgcn_wmma_f32_16x16x4_f32(false, aP, false, bmat, (short)0, c, false, false);

    const float wv = Wv[col];
    const float w0 = Wpi[col];
    const float w1 = Wpi[HIDDEN + col];
#pragma unroll
    for (int r = 0; r < 8; ++r) {
      const float hv  = fmaxf(h[r],  0.f);   // relu
      const float hpv = fmaxf(hp[r], 0.f);
      accv[r]  += hv  * wv;
      accp0[r] += hv  * w0;
      accp1[r] += hv  * w1;
      accvp[r] += hpv * wv;
    }
  }

  // Reduce over the 16 lanes of each half (masks < 16 keep halves separate).
#pragma unroll
  for (int m = 1; m < 16; m <<= 1) {
#pragma unroll
    for (int r = 0; r < 8; ++r) {
      accv[r]  += __shfl_xor(accv[r],  m, 32);
      accp0[r] += __shfl_xor(accp0[r], m, 32);
      accp1[r] += __shfl_xor(accp1[r], m, 32);
      accvp[r] += __shfl_xor(accvp[r], m, 32);
    }
  }

  if (mrow == 0) {  // lanes 0 (rows 0..7) and 16 (rows 8..15)
    const float bv0 = bv[0], bp0 = bpi[0], bp1 = bpi[1];
#pragma unroll
    for (int r = 0; r < 8; ++r) {
      const int idx = tile + half * 8 + r;
      const float v  = accv[r]  + bv0;
      const float vp = accvp[r] + bv0;
      const float l0 = accp0[r] + bp0;
      const float l1 = accp1[r] + bp1;
      const float mx = fmaxf(l0, l1);
      const float e0 = __expf(l0 - mx), e1 = __expf(l1 - mx);
      const float pia   = (act[idx] == 0 ? e0 : e1) / (e0 + e1);
      const float ratio = pia / mu[idx];
      const float rho   = fminf(1.0f, ratio);   // CLIP_RHO == CLIP_C -> cs == rhos
      const float td    = rew[idx] + GAMMA * vp * done[idx];
      out[2 * T_LEN + idx] = rho;               // rhos output
      a_arr[idx]  = GAMMA * rho;                // scan coefficient a_i = gamma*c_i
      d_arr[idx]  = rho * (td - v);             // delta_i
      v_arr[idx]  = v;
      vp_arr[idx] = vp;
    }
  }
}

// ---------------------------------------------------------------------------
// Reverse linear recurrence x_i = a_i x_{i+1} + d_i via segment composition
// (A,D): x_lo = A*x_hi + D.  compose(Left,Right) = (A_l*A_r, A_l*D_r + D_l).
// ---------------------------------------------------------------------------

// Kernel 2: per-block aggregate (each thread serially folds SEQ elements).
__global__ __launch_bounds__(SCAN_THREADS) void vtrace_block_agg(
    const float* __restrict__ a_arr, const float* __restrict__ d_arr,
    float* __restrict__ blkA, float* __restrict__ blkD)
{
  __shared__ float sA[SCAN_THREADS], sD[SCAN_THREADS];
  const int t = threadIdx.x;
  const int base = (blockIdx.x * SCAN_THREADS + t) * SEQ;
  float A = 1.f, D = 0.f;
#pragma unroll
  for (int j = SEQ - 1; j >= 0; --j) {         // prepend leftwards
    const float a = a_arr[base + j], d = d_arr[base + j];
    D = a * D + d;
    A = a * A;
  }
  sA[t] = A; sD[t] = D;
  __syncthreads();
  for (int stp = 1; stp < SCAN_THREADS; stp <<= 1) {
    const bool on = ((t & (2 * stp - 1)) == 0);
    float nA = 0.f, nD = 0.f;
    if (on) {
      const float AL = sA[t], DL = sD[t], AR = sA[t + stp], DR = sD[t + stp];
      nA = AL * AR; nD = AL * DR + DL;
    }
    __syncthreads();
    if (on) { sA[t] = nA; sD[t] = nD; }
    __syncthreads();
  }
  if (t == 0) { blkA[blockIdx.x] = sA[0]; blkD[blockIdx.x] = sD[0]; }
}

// Kernel 3: serial reverse scan over the 256 block aggregates -> per-block
// carry (value of x at the block's right boundary). Tiny; one thread.
__global__ void vtrace_carries(const float* __restrict__ blkA,
                               const float* __restrict__ blkD,
                               float* __restrict__ carry)
{
  if (threadIdx.x == 0 && blockIdx.x == 0) {
    float x = 0.f;
    for (int b = SCAN_BLOCKS - 1; b >= 0; --b) {
      carry[b] = x;
      x = blkA[b] * x + blkD[b];
    }
  }
}

// Kernel 4: per-block exclusive suffix scan of thread aggregates, then each
// thread replays its SEQ elements to emit vs and advantage.
__global__ __launch_bounds__(SCAN_THREADS) void vtrace_finalize(
    const float* __restrict__ a_arr, const float* __restrict__ d_arr,
    const float* __restrict__ v_arr, const float* __restrict__ vp_arr,
    const float* __restrict__ rew, const float* __restrict__ carry,
    float* __restrict__ out)
{
  __shared__ float sA[SCAN_THREADS], sD[SCAN_THREADS];
  const int t = threadIdx.x;
  const int base = (blockIdx.x * SCAN_THREADS + t) * SEQ;

  float la[SEQ], ld[SEQ];
#pragma unroll
  for (int j = 0; j < SEQ; ++j) { la[j] = a_arr[base + j]; ld[j] = d_arr[base + j]; }

  float A = 1.f, D = 0.f;
#pragma unroll
  for (int j = SEQ - 1; j >= 0; --j) { D = la[j] * D + ld[j]; A = la[j] * A; }
  sA[t] = A; sD[t] = D;
  __syncthreads();

  // Hillis-Steele inclusive suffix scan (compose self with right neighbor).
  for (int off = 1; off < SCAN_THREADS; off <<= 1) {
    const float ASelf = sA[t], DSelf = sD[t];
    float AR = 1.f, DR = 0.f;
    if (t + off < SCAN_THREADS) { AR = sA[t + off]; DR = sD[t + off]; }
    __syncthreads();
    sA[t] = ASelf * AR;
    sD[t] = ASelf * DR + DSelf;
    __syncthreads();
  }

  const float cb = carry[blockIdx.x];
  float eA = 1.f, eD = 0.f;                    // exclusive: composition of threads t+1..
  if (t + 1 < SCAN_THREADS) { eA = sA[t + 1]; eD = sD[t + 1]; }
  float x = eA * cb + eD;                      // ys at right boundary of this segment

#pragma unroll
  for (int j = SEQ - 1; j >= 0; --j) {
    const int i = base + j;
    const float ys_next = x;                   // ys[i+1]
    x = la[j] * x + ld[j];                     // ys[i]
    const float v = v_arr[i], vp = vp_arr[i];
    out[i]         = x + v;                                        // vs
    out[T_LEN + i] = rew[i] + GAMMA * (ys_next + vp) - v;          // advantage
  }
}

extern "C" void kernel_launch(void* const* d_in, const int* in_sizes, int n_in,
                              void* d_out, int out_size, void* d_ws, size_t ws_size,
                              hipStream_t stream) {
  (void)in_sizes; (void)n_in; (void)out_size; (void)ws_size;
  const float* s       = (const float*)d_in[0];
  const int*   act     = (const int*)  d_in[1];
  const float* rew     = (const float*)d_in[2];
  const float* s_prime = (const float*)d_in[3];
  const float* done    = (const float*)d_in[4];
  const float* mu      = (const float*)d_in[5];
  const float* W1      = (const float*)d_in[6];
  const float* b1      = (const float*)d_in[7];
  const float* Wpi     = (const float*)d_in[8];
  const float* bpi     = (const float*)d_in[9];
  const float* Wv      = (const float*)d_in[10];
  const float* bv      = (const float*)d_in[11];
  float* out = (float*)d_out;

  float* w      = (float*)d_ws;
  float* a_arr  = w;                  // T
  float* d_arr  = w + 1 * T_LEN;      // T
  float* v_arr  = w + 2 * T_LEN;      // T
  float* vp_arr = w + 3 * T_LEN;      // T
  float* blkA   = w + 4 * T_LEN;      // 256
  float* blkD   = blkA + SCAN_BLOCKS; // 256
  float* carry  = blkD + SCAN_BLOCKS; // 256

  // Phase 1: WMMA backbone + heads (128 timesteps per 256-thread block).
  vtrace_backbone_wmma<<<T_LEN / 128, 256, 0, stream>>>(
      s, act, rew, s_prime, done, mu, W1, b1, Wpi, bpi, Wv, bv,
      out, a_arr, d_arr, v_arr, vp_arr);

  // Phase 2: parallel reverse linear scan.
  vtrace_block_agg<<<SCAN_BLOCKS, SCAN_THREADS, 0, stream>>>(a_arr, d_arr, blkA, blkD);
  vtrace_carries<<<1, 32, 0, stream>>>(blkA, blkD, carry);
  vtrace_finalize<<<SCAN_BLOCKS, SCAN_THREADS, 0, stream>>>(
      a_arr, d_arr, v_arr, vp_arr, rew, carry, out);
}